// ASTRF_27135603376408
// MI455X (gfx1250) — compile-verified
//
#include <hip/hip_runtime.h>
#include <hip/hip_bf16.h>

// Problem constants (from the reference): B=4, I=16, S=4096, O=64, W=64, T=32768.
#define NB 4
#define NI 16
#define NS 4096
#define NO 64
#define NW 64
#define NT 32768

#define SCHUNK 1024          // events per workgroup chunk
#define NCHUNK (NS / SCHUNK) // 4
#define LDSWIN 8192          // f32 window: span <= 1023*7+6+64 = 7231 < 8192 (32 KB)

typedef float v2f __attribute__((ext_vector_type(2)));
typedef float v8f __attribute__((ext_vector_type(8)));

// ---------------------------------------------------------------------------
// Kernel 1: out[b,o,t] = bias[o]  (float4 vectorized; T = 2^15)
// ---------------------------------------------------------------------------
__global__ void astrf_bias_init(const float* __restrict__ bias,
                                float* __restrict__ out) {
    size_t id = (size_t)blockIdx.x * blockDim.x + threadIdx.x;  // one float4
    size_t e  = id * 4;
    int o = (int)((e >> 15) & (NO - 1));                        // (e / T) % O
    float bv = bias[o];
    *(float4*)(out + e) = make_float4(bv, bv, bv, bv);
}

// ---------------------------------------------------------------------------
// Kernel 2: fused GEMM + overlap-add.
//   - workgroup = (b, o, s-chunk of 1024 events); 8 waves, 256 threads
//   - wave computes 16x16 trf tiles with 4 chained V_WMMA_F32_16X16X4_F32
//   - scatter-adds into a 32 KB LDS time window via ds_add_f32 (WGP-local)
//   - flushes only the live span to global with global_atomic_add_f32
//     (chunk halos overlap between neighboring workgroups -> atomics needed)
// ---------------------------------------------------------------------------
__global__ void astrf_wmma_lds(const float* __restrict__ x,      // (B,I,S)
                               const float* __restrict__ wgt,    // (O,I,W)
                               const int*   __restrict__ srcIdx, // (B,S) sorted
                               float*       __restrict__ out) {  // (B,O,T)
    __shared__ float acc[LDSWIN];

    int wg = blockIdx.x;
    const int ch = wg & (NCHUNK - 1);  wg >>= 2;  // s-chunk
    const int o  = wg & (NO - 1);      wg >>= 6;  // out channel
    const int b  = wg;                            // batch

    const int s0   = ch * SCHUNK;
    const int base = srcIdx[b * NS + s0];         // min index of chunk (sorted)

    // zero the accumulation window
    for (int i = threadIdx.x; i < LDSWIN; i += 256) acc[i] = 0.0f;
    __syncthreads();

    const int lane = threadIdx.x & 31;
    const int wave = threadIdx.x >> 5;
    const int m    = lane & 15;   // A-row (=w offset) / B-col (=s offset)
    const int hi   = lane >> 4;   // lane half selects K sub-pair

    // wave -> fixed w-tile (A operands loop-invariant), half of the s-tiles
    const int wt     = wave & 3;        // w-tile 0..3
    const int sgroup = wave >> 2;       // 0/1: s-tile parity
    const int w0     = wt * 16;

    // Hoisted A operands: A[m,k] = wgt[o, k, w0+m], k = kk*4 + 2*hi + {0,1}
    const float* wp = wgt + (size_t)o * (NI * NW) + (w0 + m);
    v2f a[4];
#pragma unroll
    for (int kk = 0; kk < 4; ++kk) {
        const int k0 = kk * 4 + 2 * hi;
        a[kk].x = wp[(size_t)(k0)     * NW];
        a[kk].y = wp[(size_t)(k0 + 1) * NW];
    }

    const float* xb   = x + (size_t)b * (NI * NS);
    const int*   sidx = srcIdx + b * NS;

    // 64 s-tiles per chunk, 2 waves share them (sgroup + 2*t)
    for (int t = 0; t < SCHUNK / 16 / 2; ++t) {
        const int stile = sgroup + 2 * t;
        const int ss    = s0 + stile * 16;

        // B[k,n] = x[b, k, ss+n], n == m
        const float* xp = xb + (ss + m);
        v8f c = {};
#pragma unroll
        for (int kk = 0; kk < 4; ++kk) {
            const int k0 = kk * 4 + 2 * hi;
            v2f bv;
            bv.x = xp[(size_t)(k0)     * NS];
            bv.y = xp[(size_t)(k0 + 1) * NS];
            c = __builtin_amdgcn_wmma_f32_16x16x4_f32(
                    false, a[kk], false, bv, (short)0, c, false, false);
        }

        // lane holds event ss+m; C element r is w = w0 + 8*hi + r
        const int toff = sidx[ss + m] - base + w0 + 8 * hi;  // < 7231+? < LDSWIN
#pragma unroll
        for (int r = 0; r < 8; ++r) {
            __hip_atomic_fetch_add(&acc[toff + r], c[r], __ATOMIC_RELAXED,
                                   __HIP_MEMORY_SCOPE_WORKGROUP);
        }
    }
    __syncthreads();

    // flush live span [base, base+span) -> global (halo overlap => atomics)
    const int span = sidx[s0 + SCHUNK - 1] + NW - base;      // <= 7231
    float* op = out + (size_t)(b * NO + o) * NT + base;
    for (int i = threadIdx.x; i < span; i += 256) {
        __hip_atomic_fetch_add(op + i, acc[i], __ATOMIC_RELAXED,
                               __HIP_MEMORY_SCOPE_AGENT);
    }
}

// ---------------------------------------------------------------------------
extern "C" void kernel_launch(void* const* d_in, const int* in_sizes, int n_in,
                              void* d_out, int out_size, void* d_ws, size_t ws_size,
                              hipStream_t stream) {
    const float* x      = (const float*)d_in[0];  // (B,I,S)
    const float* weight = (const float*)d_in[1];  // (O,I,W)
    const float* bias   = (const float*)d_in[2];  // (O,)
    const int*   srcIdx = (const int*)d_in[3];    // (B,S) int32
    float*       out    = (float*)d_out;          // (B,O,T)

    // 1) out = bias broadcast (B*O*T/4 float4 stores)
    const int initQuads = NB * NO * NT / 4;       // 2,097,152
    astrf_bias_init<<<initQuads / 256, 256, 0, stream>>>(bias, out);

    // 2) fused WMMA + LDS overlap-add + atomic flush
    const int nWG = NB * NO * NCHUNK;             // 1024 workgroups
    astrf_wmma_lds<<<nWG, 256, 0, stream>>>(x, weight, srcIdx, out);
}